// Model_29944511987736
// MI455X (gfx1250) — compile-verified
//
#include <hip/hip_runtime.h>
#include <hip/hip_bf16.h>

// MI455X (gfx1250) kernel for the Scallop diffminmaxprob fixpoint.
//
// Reference analysis:
//  - The RandNet matmul/softmax path is dead code (output unused) -> skipped.
//  - Live output: M[t0,t1,v0,v2] (16^4 floats), least fixpoint of
//      rule1: M[t,t]      = S[t]
//      rule2: M[t,t]     |= M[t,t] (max-min∘) S[t]
//      rule3: M[t0,t1]   |= M[t0,t1-1] (max-min∘) S[t1]
//    Closed form (reached exactly by the reference's 40 iterations):
//      M[t0,t0]   = closure(S[t0])   (max-min path closure, paths <= 16)
//      M[t0,t1>t0]= closure(S[t0]) ∘ S[t0+1] ∘ ... ∘ S[t1]
//      M[t0,t1<t0]= 0
//    max/min are selections -> bitwise identical to the iterated version.
//
//  - One workgroup per t0 (16 blocks x 256 threads). All state in LDS.
//  - S (16 KB) staged via CDNA5 async global->LDS engine (ASYNCcnt) when the
//    builtin exists; vector-load fallback otherwise.

#define TS 16

#if defined(__gfx1250__) && __has_builtin(__builtin_amdgcn_global_load_async_to_lds_b128)
#define USE_ASYNC_LDS 1
#else
#define USE_ASYNC_LDS 0
#endif

__global__ __launch_bounds__(256) void scallop_fixpoint_kernel(
    const float* __restrict__ S,   // [16][16][16] single_match
    float* __restrict__ out) {     // [16][16][16][16] M
  __shared__ __align__(16) float Ss[TS * TS * TS];  // 16 KB: all S[t][v0][v1]
  __shared__ float A[2][TS * TS];                   // ping-pong product matrix

  const int tid = threadIdx.x;   // 256 threads: tid = v0*16 + v2
  const int t0  = blockIdx.x;    // 16 blocks
  const int v0  = tid >> 4;
  const int v2  = tid & 15;

  // ---- Stage S into LDS (4096 floats = 1024 x b128; 4 per thread) ----
#if USE_ASYNC_LDS
  {
    typedef int v4i __attribute__((vector_size(16)));
    typedef __attribute__((address_space(1))) v4i GV4;  // global-side v4i
    typedef __attribute__((address_space(3))) v4i LV4;  // LDS-side v4i
#pragma unroll
    for (int i = 0; i < 4; ++i) {
      const int id = i * 256 + tid;  // float4 granule; lanes contiguous
      __builtin_amdgcn_global_load_async_to_lds_b128(
          (GV4*)(S + id * 4), (LV4*)&Ss[id * 4], /*offset=*/0, /*cpol=*/0);
    }
    asm volatile("s_wait_asynccnt 0" ::: "memory");
  }
#else
#pragma unroll
  for (int i = 0; i < 4; ++i) {
    const int id = i * 256 + tid;
    reinterpret_cast<float4*>(Ss)[id] =
        reinterpret_cast<const float4*>(S)[id];
  }
#endif
  __syncthreads();

  // ---- A = S[t0] ----
  A[0][tid] = Ss[t0 * 256 + tid];
  __syncthreads();

  // ---- Max-min closure of A: 5 doubling steps (covers path length <= 32) ----
  int cur = 0;
#pragma unroll
  for (int it = 0; it < 5; ++it) {
    const float* Ac = A[cur];
    float best = fminf(Ac[v0 * 16 + 0], Ac[0 * 16 + v2]);
#pragma unroll
    for (int v1 = 1; v1 < TS; ++v1)
      best = fmaxf(best, fminf(Ac[v0 * 16 + v1], Ac[v1 * 16 + v2]));
    const float nv = fmaxf(Ac[tid], best);
    __syncthreads();               // all reads of A[cur] done
    A[cur ^ 1][tid] = nv;
    __syncthreads();
    cur ^= 1;
  }

  // ---- Emit band t1 < t0 (zeros) and the diagonal t1 == t0 (closure) ----
  for (int t1 = 0; t1 < t0; ++t1)
    out[(t0 * TS + t1) * 256 + tid] = 0.0f;
  out[(t0 * TS + t0) * 256 + tid] = A[cur][tid];

  // ---- Chain: P <- P ∘ S[t1], emit M[t0,t1] for t1 = t0+1 .. 15 ----
  for (int t1 = t0 + 1; t1 < TS; ++t1) {
    const float* P  = A[cur];
    const float* Sp = &Ss[t1 * 256];
    float best = fminf(P[v0 * 16 + 0], Sp[0 * 16 + v2]);
#pragma unroll
    for (int v1 = 1; v1 < TS; ++v1)
      best = fmaxf(best, fminf(P[v0 * 16 + v1], Sp[v1 * 16 + v2]));
    __syncthreads();               // all reads of A[cur] done
    A[cur ^ 1][tid] = best;
    out[(t0 * TS + t1) * 256 + tid] = best;
    __syncthreads();
    cur ^= 1;
  }
}

extern "C" void kernel_launch(void* const* d_in, const int* in_sizes, int n_in,
                              void* d_out, int out_size, void* d_ws, size_t ws_size,
                              hipStream_t stream) {
  // setup_inputs() order: x, fc1_w, fc1_b, fc2_w, fc2_b, single_match
  const float* single_match = (const float*)d_in[5];
  float* out = (float*)d_out;  // 16^4 = 65536 floats, all written below

  scallop_fixpoint_kernel<<<dim3(TS), dim3(256), 0, stream>>>(single_match, out);
}